// LocalFeatureEnrichment_56831007261016
// MI455X (gfx1250) — compile-verified
//
#include <hip/hip_runtime.h>
#include <hip/hip_bf16.h>

// Problem constants (from reference): B=16, N=1024, D=1024, P=256, K=16
#define BB   16
#define NN   1024
#define DD   1024
#define PP   256
#define KNN  16
#define MM   (BB * NN)          // 16384 total points

typedef float v2f __attribute__((ext_vector_type(2)));
typedef float v8f __attribute__((ext_vector_type(8)));

__device__ __forceinline__ v8f wmma4(v2f a, v2f b, v8f c) {
  // V_WMMA_F32_16X16X4_F32 : D = A(16x4) * B(4x16) + C  (fp32 throughout)
  return __builtin_amdgcn_wmma_f32_16x16x4_f32(
      /*neg_a=*/false, a, /*neg_b=*/false, b,
      /*c_mod=*/(short)0, c, /*reuse_a=*/false, /*reuse_b=*/false);
}

// ---------------------------------------------------------------------------
// K2: exact-KNN (K=16) per point. One block = 256 query points of one batch.
// Batch's 1024 points + squared norms staged in LDS; per-thread top-16 kept
// in LDS (dynamic-index friendly). d2 = sq_n + sq_m - 2*dot (matches ref).
// ---------------------------------------------------------------------------
__global__ __launch_bounds__(256) void knn_kernel(const float* __restrict__ pts,
                                                  int* __restrict__ idx) {
  __shared__ float px[NN], py[NN], pz[NN], sq[NN];
  __shared__ float bd[256][KNN];
  __shared__ int   bi[256][KNN];
  const int b = blockIdx.x >> 2;          // 4 blocks per batch
  const int chunk = blockIdx.x & 3;
  const int t = threadIdx.x;
  const float* pb = pts + (size_t)b * NN * 3;
  for (int i = t; i < NN; i += 256) {
    float x = pb[i * 3 + 0], y = pb[i * 3 + 1], z = pb[i * 3 + 2];
    px[i] = x; py[i] = y; pz[i] = z;
    sq[i] = x * x + y * y + z * z;
  }
  __syncthreads();
  const int n = chunk * 256 + t;
  const float qx = px[n], qy = py[n], qz = pz[n], qs = sq[n];
  float* d = bd[t];
  int*  ii = bi[t];
#pragma unroll
  for (int j = 0; j < KNN; ++j) { d[j] = 3.4e38f; ii[j] = 0; }
  float worst = 3.4e38f;
  int wslot = 0;
  for (int m = 0; m < NN; ++m) {
    float d2 = qs + sq[m] - 2.0f * (qx * px[m] + qy * py[m] + qz * pz[m]);
    if (d2 < worst) {                     // strict '<' keeps earliest index on ties
      d[wslot] = d2; ii[wslot] = m;
      worst = d[0]; wslot = 0;
#pragma unroll
      for (int j = 1; j < KNN; ++j)
        if (d[j] > worst) { worst = d[j]; wslot = j; }
    }
  }
  int* op = idx + ((size_t)b * NN + n) * KNN;
#pragma unroll
  for (int j = 0; j < KNN; ++j) op[j] = ii[j];
}

// ---------------------------------------------------------------------------
// K1: F = relu(features[M,1024] @ W_down[1024,256] + b_down)
// One wave -> 16x32 output tile (two 16x16 WMMA accumulators).
// ---------------------------------------------------------------------------
__global__ __launch_bounds__(256) void gemm_down_kernel(const float* __restrict__ X,
                                                        const float* __restrict__ W,
                                                        const float* __restrict__ bias,
                                                        float* __restrict__ F) {
  const int wave = (blockIdx.x * blockDim.x + threadIdx.x) >> 5;   // 8192 waves
  const int lane = threadIdx.x & 31;
  const int h = lane >> 4, l = lane & 15;
  const int r0 = (wave >> 3) << 4;     // M tile: 1024 tiles
  const int c0 = (wave & 7) << 5;      // N tile pair: 8 pairs of 32 cols
  v8f acc0 = {}; v8f acc1 = {};
  const float* xrow = X + (size_t)(r0 + l) * DD;
  for (int k = 0; k < DD; k += 4) {
    const int kb = k + 2 * h;
    v2f a = *(const v2f*)(xrow + kb);                    // A[M=l][K=kb,kb+1]
    v2f b0, b1;
    b0.x = W[(size_t)kb * PP + c0 + l];
    b0.y = W[(size_t)(kb + 1) * PP + c0 + l];
    b1.x = W[(size_t)kb * PP + c0 + 16 + l];
    b1.y = W[(size_t)(kb + 1) * PP + c0 + 16 + l];
    acc0 = wmma4(a, b0, acc0);
    acc1 = wmma4(a, b1, acc1);
  }
  const float bia0 = bias[c0 + l], bia1 = bias[c0 + 16 + l];
#pragma unroll
  for (int v = 0; v < 8; ++v) {
    const int row = r0 + v + 8 * h;
    F[(size_t)row * PP + c0 + l]      = fmaxf(acc0[v] + bia0, 0.0f);
    F[(size_t)row * PP + c0 + 16 + l] = fmaxf(acc1[v] + bia1, 0.0f);
  }
}

// ---------------------------------------------------------------------------
// K3: Acen = (F @ (W1-W2))*sc + sh ;  Bnbr = (F @ W2)*sc
//   with sc = gamma*rsqrt(var+eps), sh = (b_edge - mean)*sc + beta  (BN folded)
// W1 = W_edge[0:256], W2 = W_edge[256:512]  (row-major [512,256]).
// One wave -> one 16x16 tile of BOTH outputs (shared A fragment).
// ---------------------------------------------------------------------------
__global__ __launch_bounds__(256) void gemm_edge_kernel(const float* __restrict__ F,
                                                        const float* __restrict__ We,
                                                        const float* __restrict__ b_edge,
                                                        const float* __restrict__ gamma,
                                                        const float* __restrict__ beta,
                                                        const float* __restrict__ mean,
                                                        const float* __restrict__ var,
                                                        float* __restrict__ Acen,
                                                        float* __restrict__ Bnbr) {
  const int wave = (blockIdx.x * blockDim.x + threadIdx.x) >> 5;   // 16384 waves
  const int lane = threadIdx.x & 31;
  const int h = lane >> 4, l = lane & 15;
  const int r0 = (wave >> 4) << 4;     // 1024 M tiles
  const int c0 = (wave & 15) << 4;     // 16 N tiles
  v8f accA = {}; v8f accB = {};
  const float* frow = F + (size_t)(r0 + l) * PP;
  for (int k = 0; k < PP; k += 4) {
    const int kb = k + 2 * h;
    v2f a = *(const v2f*)(frow + kb);
    v2f w1, w2;
    w1.x = We[(size_t)kb * PP + c0 + l];
    w1.y = We[(size_t)(kb + 1) * PP + c0 + l];
    w2.x = We[(size_t)(PP + kb) * PP + c0 + l];
    w2.y = We[(size_t)(PP + kb + 1) * PP + c0 + l];
    v2f wd = w1 - w2;
    accA = wmma4(a, wd, accA);
    accB = wmma4(a, w2, accB);
  }
  const int c = c0 + l;
  const float sc = gamma[c] * rsqrtf(var[c] + 1e-5f);
  const float sh = (b_edge[c] - mean[c]) * sc + beta[c];
#pragma unroll
  for (int v = 0; v < 8; ++v) {
    const int row = r0 + v + 8 * h;
    Acen[(size_t)row * PP + c] = accA[v] * sc + sh;
    Bnbr[(size_t)row * PP + c] = accB[v] * sc;
  }
}

// ---------------------------------------------------------------------------
// K4: local[n,p] = relu( max_k ( Acen[n,p] + Bnbr[b*1024 + idx[n,k], p] ) )
// One block per point; 256 threads over channels -> fully coalesced gathers
// (each gathered row is 1KB; per-batch Bnbr slab is 1MB -> L2 resident).
// ---------------------------------------------------------------------------
__global__ __launch_bounds__(256) void aggregate_kernel(const float* __restrict__ Acen,
                                                        const float* __restrict__ Bnbr,
                                                        const int* __restrict__ idx,
                                                        float* __restrict__ localf) {
  const int n = blockIdx.x;            // 0..16383
  const int base = (n >> 10) << 10;    // batch * 1024
  const int p = threadIdx.x;
  const int* ip = idx + (size_t)n * KNN;
  const float a = Acen[(size_t)n * PP + p];
  float mx = -3.4e38f;
#pragma unroll
  for (int k = 0; k < KNN; ++k) {
    const int g = base + ip[k];
    mx = fmaxf(mx, a + Bnbr[(size_t)g * PP + p]);
  }
  localf[(size_t)n * PP + p] = fmaxf(mx, 0.0f);   // relu after max (relu monotone)
}

// ---------------------------------------------------------------------------
// K5: out = features + local[M,256] @ W_up[256,1024] + b_up
// One wave -> 16x32 output tile.
// ---------------------------------------------------------------------------
__global__ __launch_bounds__(256) void gemm_up_kernel(const float* __restrict__ L,
                                                      const float* __restrict__ W,
                                                      const float* __restrict__ b_up,
                                                      const float* __restrict__ feats,
                                                      float* __restrict__ out) {
  const int wave = (blockIdx.x * blockDim.x + threadIdx.x) >> 5;   // 32768 waves
  const int lane = threadIdx.x & 31;
  const int h = lane >> 4, l = lane & 15;
  const int r0 = (wave >> 5) << 4;     // 1024 M tiles
  const int c0 = (wave & 31) << 5;     // 32 pairs of 32 cols
  v8f acc0 = {}; v8f acc1 = {};
  const float* lrow = L + (size_t)(r0 + l) * PP;
  for (int k = 0; k < PP; k += 4) {
    const int kb = k + 2 * h;
    v2f a = *(const v2f*)(lrow + kb);
    v2f b0, b1;
    b0.x = W[(size_t)kb * DD + c0 + l];
    b0.y = W[(size_t)(kb + 1) * DD + c0 + l];
    b1.x = W[(size_t)kb * DD + c0 + 16 + l];
    b1.y = W[(size_t)(kb + 1) * DD + c0 + 16 + l];
    acc0 = wmma4(a, b0, acc0);
    acc1 = wmma4(a, b1, acc1);
  }
  const float bia0 = b_up[c0 + l], bia1 = b_up[c0 + 16 + l];
#pragma unroll
  for (int v = 0; v < 8; ++v) {
    const int row = r0 + v + 8 * h;
    const size_t o0 = (size_t)row * DD + c0 + l;
    const size_t o1 = (size_t)row * DD + c0 + 16 + l;
    out[o0] = acc0[v] + bia0 + feats[o0];
    out[o1] = acc1[v] + bia1 + feats[o1];
  }
}

extern "C" void kernel_launch(void* const* d_in, const int* in_sizes, int n_in,
                              void* d_out, int out_size, void* d_ws, size_t ws_size,
                              hipStream_t stream) {
  const float* points   = (const float*)d_in[0];
  const float* features = (const float*)d_in[1];
  const float* W_down   = (const float*)d_in[2];
  const float* b_down   = (const float*)d_in[3];
  const float* W_edge   = (const float*)d_in[4];
  const float* b_edge   = (const float*)d_in[5];
  const float* bn_gamma = (const float*)d_in[6];
  const float* bn_beta  = (const float*)d_in[7];
  const float* bn_mean  = (const float*)d_in[8];
  const float* bn_var   = (const float*)d_in[9];
  const float* W_up     = (const float*)d_in[10];
  const float* b_up     = (const float*)d_in[11];
  float* out = (float*)d_out;

  // Workspace layout: F(16MB) | Acen(16MB) | Bnbr(16MB) | idx(1MB); local reuses F.
  float* F    = (float*)d_ws;
  float* Acen = F    + (size_t)MM * PP;
  float* Bnbr = Acen + (size_t)MM * PP;
  int*   idx  = (int*)(Bnbr + (size_t)MM * PP);
  float* localf = F;

  knn_kernel<<<BB * (NN / 256), 256, 0, stream>>>(points, idx);
  gemm_down_kernel<<<(MM / 16) * (PP / 32) / 8, 256, 0, stream>>>(features, W_down, b_down, F);
  gemm_edge_kernel<<<(MM / 16) * (PP / 16) / 8, 256, 0, stream>>>(
      F, W_edge, b_edge, bn_gamma, bn_beta, bn_mean, bn_var, Acen, Bnbr);
  aggregate_kernel<<<MM, 256, 0, stream>>>(Acen, Bnbr, idx, localf);
  gemm_up_kernel<<<(MM / 16) * (DD / 32) / 8, 256, 0, stream>>>(localf, W_up, b_up, features, out);
}